// MetaPathEncoder_3444563771401
// MI455X (gfx1250) — compile-verified
//
#include <hip/hip_runtime.h>

#define N_NODES 10000
#define N_PATHS 4
#define N_EDGES 160000
#define IN_DIM  512
#define OUT_DIM 512
#define K_TOT   (N_PATHS * IN_DIM)   // 2048

typedef float v2f __attribute__((ext_vector_type(2)));
typedef float v8f __attribute__((ext_vector_type(8)));

// Workspace layout (in floats):
//  [0, 40000)                        deg_out_inv[p][n]  (counts, then rsqrt in place)
//  [40000, 80000)                    deg_in_inv[p][n]
//  [80000, 80000 + 4*10000*512)      agg[p][n][c]  == fully normalized A matrix
#define WS_DEGOUT 0
#define WS_DEGIN  (N_PATHS * N_NODES)
#define WS_AGG    (2 * N_PATHS * N_NODES)
#define WS_TOTAL  (WS_AGG + N_PATHS * N_NODES * IN_DIM)   // 20,560,000 floats

// ---------------------------------------------------------------- zero ws
__global__ __launch_bounds__(256)
void mp_zero_ws(float* __restrict__ ws, int total) {
    int i4 = (blockIdx.x * 256 + threadIdx.x) * 4;
    if (i4 + 3 < total) {
        *(float4*)(ws + i4) = make_float4(0.f, 0.f, 0.f, 0.f);
    } else {
        for (int i = i4; i < total; ++i) ws[i] = 0.f;
    }
}

// ---------------------------------------------------------------- degrees
__global__ __launch_bounds__(256)
void mp_degrees(const long long* __restrict__ src,
                const long long* __restrict__ dst,
                float* __restrict__ ws) {
    int tid = blockIdx.x * 256 + threadIdx.x;
    if (tid >= N_PATHS * N_EDGES) return;
    int p = tid / N_EDGES;
    int s = (int)src[tid];
    int d = (int)dst[tid];
    atomicAdd(ws + WS_DEGOUT + p * N_NODES + s, 1.0f);
    atomicAdd(ws + WS_DEGIN  + p * N_NODES + d, 1.0f);
}

// ---------------------------------------------------------------- rsqrt(max(deg,1))
__global__ __launch_bounds__(256)
void mp_rsqrt(float* __restrict__ ws) {
    int tid = blockIdx.x * 256 + threadIdx.x;
    if (tid >= 2 * N_PATHS * N_NODES) return;
    ws[tid] = rsqrtf(fmaxf(ws[tid], 1.0f));
}

// ---------------------------------------------------------------- scatter SpMM
// 128 threads per (path, edge); each thread moves a float4 column chunk.
// Both normalizations fused: agg[p][d] += feat[s] * degout^-1/2[s] * degin^-1/2[d]
__global__ __launch_bounds__(256)
void mp_scatter(const float* __restrict__ feat,
                const long long* __restrict__ src,
                const long long* __restrict__ dst,
                float* __restrict__ ws) {
    int tid = blockIdx.x * 256 + threadIdx.x;            // < 4*160000*128 = 81.92M
    int chunk = tid & 127;
    int eid   = tid >> 7;                                // path*E + e
    int p     = eid / N_EDGES;
    int s = (int)src[eid];
    int d = (int)dst[eid];
    float scale = ws[WS_DEGOUT + p * N_NODES + s] * ws[WS_DEGIN + p * N_NODES + d];
    float4 v = *(const float4*)(feat + (size_t)s * IN_DIM + chunk * 4);
    float* ag = ws + WS_AGG + ((size_t)(p * N_NODES + d)) * IN_DIM + chunk * 4;
    atomicAdd(ag + 0, v.x * scale);
    atomicAdd(ag + 1, v.y * scale);
    atomicAdd(ag + 2, v.z * scale);
    atomicAdd(ag + 3, v.w * scale);
}

// ---------------------------------------------------------------- fused GEMM
// out[n][o] = 0.25 * sum_k A[n][k] * B[k][o] + 0.25 * sum_p b[p][o]
// A[n][p*512+c] = agg[p][n][c] (pre-normalized), B[p*512+c][o] = W[p][c][o]
// Block 256 thr = 8 waves (2M x 4N); wave tile 32x32 (4 WMMA accumulators).
// Block tile 64(M) x 128(N), BK=32; async global->LDS staging, double-buffered.
#define AS_STRIDE 36                 // floats; 144B rows (16B aligned, conflict-free)
#define BS_STRIDE 136                // floats; 544B rows
#define AS_FLOATS (64 * AS_STRIDE)   // 2304
#define BS_FLOATS (32 * BS_STRIDE)   // 4352

__global__ __launch_bounds__(256)
void mp_gemm_wmma(const float* __restrict__ aggA,
                  const float* __restrict__ W,
                  const float* __restrict__ b,
                  float* __restrict__ out) {
    __shared__ __align__(16) float smem[2 * (AS_FLOATS + BS_FLOATS)];

    const int tid  = threadIdx.x;
    const int lane = tid & 31;
    const int wid  = tid >> 5;      // 0..7
    const int wm   = wid & 1;       // M wave tile (32 rows each)
    const int wn   = wid >> 1;      // N wave tile (32 cols each, 0..3)
    const int half = lane >> 4;     // ISA A/B frag lane group
    const int lcol = lane & 15;

    const int mBlock = blockIdx.y * 64;
    const int nBlock = blockIdx.x * 128;

    // LDS byte offset of smem (strip flat aperture via addrspace(3) cast)
    typedef __attribute__((address_space(3))) float lds_f;
    const unsigned smem_base = (unsigned)(unsigned long long)(lds_f*)smem;

    // staging maps
    const int ar  = tid >> 3;          // A row 0..31 (+32 second pass)
    const int af  = (tid & 7) << 2;    // A float offset in 32-float k chunk
    const int bkr = tid >> 3;          // B k row 0..31
    const int bnf = (tid & 7) << 4;    // B col offset (16 floats = 4 x b128)

    auto stage = [&](int k0, int buf) {
        const int p     = k0 >> 9;     // 512 % 32 == 0: one path per BK chunk
        const int cbase = k0 & 511;
        // A tile: 64 x 32, two b128 per thread (row clamped; garbage rows are
        // only ever multiplied into out rows >= N_NODES which the store guards)
        #pragma unroll
        for (int rr = 0; rr < 64; rr += 32) {
            const int r = ar + rr;
            int n = mBlock + r;
            if (n >= N_NODES) n = N_NODES - 1;
            const float* g = aggA + ((size_t)(p * N_NODES + n)) * IN_DIM + cbase + af;
            unsigned l = smem_base +
                (unsigned)((buf ? AS_FLOATS : 0) + r * AS_STRIDE + af) * 4u;
            asm volatile("global_load_async_to_lds_b128 %0, %1, off"
                         :: "v"(l), "v"(g) : "memory");
        }
        // B tile: 32 x 128, four b128 per thread
        const float* g0 = W + ((size_t)(p * IN_DIM + cbase + bkr)) * OUT_DIM + nBlock + bnf;
        unsigned lb = smem_base +
            (unsigned)(2 * AS_FLOATS + (buf ? BS_FLOATS : 0) + bkr * BS_STRIDE + bnf) * 4u;
        #pragma unroll
        for (int j = 0; j < 4; ++j) {
            asm volatile("global_load_async_to_lds_b128 %0, %1, off"
                         :: "v"(lb + 16u * j), "v"(g0 + 4 * j) : "memory");
        }
    };

    v8f acc00 = {}, acc01 = {}, acc10 = {}, acc11 = {};

    stage(0, 0);
    const int NCHUNK = K_TOT / 32;     // 64
    for (int c = 0; c < NCHUNK; ++c) {
        asm volatile("s_wait_asynccnt 0" ::: "memory");  // our stage(c) landed
        __syncthreads();                                  // everyone's landed; prev reads done
        if (c + 1 < NCHUNK) stage((c + 1) * 32, (c + 1) & 1);

        const float* A = smem + ((c & 1) ? AS_FLOATS : 0);
        const float* B = smem + 2 * AS_FLOATS + ((c & 1) ? BS_FLOATS : 0);

        #pragma unroll
        for (int kk = 0; kk < 32; kk += 4) {
            // A frags: lane<16 -> (K=kk,kk+1), lane>=16 -> (kk+2,kk+3); M row = lcol
            v2f a0, a1, b0, b1;
            const float* ap0 = A + (wm * 32 + lcol) * AS_STRIDE + kk + 2 * half;
            const float* ap1 = ap0 + 16 * AS_STRIDE;
            a0.x = ap0[0]; a0.y = ap0[1];
            a1.x = ap1[0]; a1.y = ap1[1];
            // B frags mirrored over N = lcol
            const float* bp = B + (kk + 2 * half) * BS_STRIDE + wn * 32 + lcol;
            b0.x = bp[0];  b0.y = bp[BS_STRIDE];
            b1.x = bp[16]; b1.y = bp[BS_STRIDE + 16];

            acc00 = __builtin_amdgcn_wmma_f32_16x16x4_f32(false, a0, false, b0, (short)0, acc00, false, false);
            acc01 = __builtin_amdgcn_wmma_f32_16x16x4_f32(false, a0, false, b1, (short)0, acc01, false, false);
            acc10 = __builtin_amdgcn_wmma_f32_16x16x4_f32(false, a1, false, b0, (short)0, acc10, false, false);
            acc11 = __builtin_amdgcn_wmma_f32_16x16x4_f32(false, a1, false, b1, (short)0, acc11, false, false);
        }
        // no trailing barrier needed: next iteration's top barrier orders
        // buffer reuse, and stage(c+1) writes only the other buffer.
    }

    // Epilogue: 0.25 * acc + mean bias. C/D: VGPR r -> M = half*8 + r, N = lcol.
    #pragma unroll
    for (int ni = 0; ni < 2; ++ni) {
        const int col = nBlock + wn * 32 + ni * 16 + lcol;
        const float bias = 0.25f * (b[col] + b[OUT_DIM + col] +
                                    b[2 * OUT_DIM + col] + b[3 * OUT_DIM + col]);
        #pragma unroll
        for (int mi = 0; mi < 2; ++mi) {
            const v8f acc = (mi == 0) ? (ni == 0 ? acc00 : acc01)
                                      : (ni == 0 ? acc10 : acc11);
            const int m0 = mBlock + wm * 32 + mi * 16 + half * 8;
            #pragma unroll
            for (int r = 0; r < 8; ++r) {
                const int row = m0 + r;
                if (row < N_NODES)
                    out[(size_t)row * OUT_DIM + col] = 0.25f * acc[r] + bias;
            }
        }
    }
}

// ---------------------------------------------------------------- launch
extern "C" void kernel_launch(void* const* d_in, const int* in_sizes, int n_in,
                              void* d_out, int out_size, void* d_ws, size_t ws_size,
                              hipStream_t stream) {
    const float*     feat = (const float*)d_in[0];
    const long long* src  = (const long long*)d_in[1];   // int64 indices
    const long long* dst  = (const long long*)d_in[2];
    const float*     W    = (const float*)d_in[3];
    const float*     b    = (const float*)d_in[4];
    float* out = (float*)d_out;
    float* ws  = (float*)d_ws;

    (void)in_sizes; (void)n_in; (void)out_size; (void)ws_size;

    // 1) zero degree + agg scratch
    {
        const int total = WS_TOTAL;
        const int n4 = (total + 3) / 4;
        mp_zero_ws<<<(n4 + 255) / 256, 256, 0, stream>>>(ws, total);
    }
    // 2) degree counts
    mp_degrees<<<(N_PATHS * N_EDGES + 255) / 256, 256, 0, stream>>>(src, dst, ws);
    // 3) rsqrt(clip(deg,1))
    mp_rsqrt<<<(2 * N_PATHS * N_NODES + 255) / 256, 256, 0, stream>>>(ws);
    // 4) scatter-add SpMM with both degree normalizations fused
    {
        const long long work = (long long)N_PATHS * N_EDGES * 128;   // 81.92M threads
        const int blocks = (int)(work / 256);
        mp_scatter<<<blocks, 256, 0, stream>>>(feat, src, dst, ws);
    }
    // 5) fused f32 WMMA GEMM (async double-buffered staging) + mean + bias
    {
        dim3 grid(OUT_DIM / 128, (N_NODES + 63) / 64);   // 4 x 157
        mp_gemm_wmma<<<grid, 256, 0, stream>>>(ws + WS_AGG, W, b, out);
    }
}